// Preprocess_73065983639768
// MI455X (gfx1250) — compile-verified
//
#include <hip/hip_runtime.h>
#include <hip/hip_bf16.h>

typedef float v2f __attribute__((ext_vector_type(2)));
typedef float v8f __attribute__((ext_vector_type(8)));

#define OUTW    256
#define PADW    316        // 256 + 2*30
#define NIMG    48         // 16*3
#define NTAP    127
#define FOFF    16         // filter taps live at fpad[16 .. 143); zeros elsewhere
#define FPADN   176
#define REXTW   400        // cols = 49 + s, s in [-33, 351); zero outside [0,316)
#define RCOFF   49         // chosen so (RCOFF-33)=16 keeps A loads 8B-aligned
#define HEXTR   384        // rows = 33 + py; zero aprons [0,33) and [349,384)
#define HROFF   33

__device__ __forceinline__ int reflect256(int j) {
    // jnp.pad 'reflect' (numpy, no edge repeat); pad < 256 so one bounce
    j = (j < 0) ? -j : j;
    return (j > 255) ? (510 - j) : j;
}

// ---- padded filter table: fpad[FOFF+t] = exp2(-((t-63)/21)^2)/sum, else 0 ----
__global__ void Pre_init_filter(float* __restrict__ fpad) {
    int t = threadIdx.x;                      // 192 threads
    float sum = 0.f;
    for (int i = 0; i < NTAP; ++i) {
        float x = (float)(i - 63) * (1.0f / 21.0f);
        sum += exp2f(-x * x);
    }
    if (t < FPADN) {
        float v = 0.f;
        if (t >= FOFF && t < FOFF + NTAP) {
            float x = (float)(t - FOFF - 63) * (1.0f / 21.0f);
            v = exp2f(-x * x) / sum;
        }
        fpad[t] = v;
    }
}

// ---- fused downsample + reflect/zero-padded row image --------------------
// Rext[img][r][RCOFF+s] = (0<=s<316) ? R[img][r][reflect(s-30)] : 0
// R[r][c] = 0.25*(I[4r+1,4c+1]+I[4r+1,4c+2]+I[4r+2,4c+1]+I[4r+2,4c+2])
__global__ void Pre_build_rext(const float* __restrict__ I, float* __restrict__ Rext) {
    int row = blockIdx.x;                     // 48*256 rows
    int img = row >> 8;
    int r   = row & 255;
    const float* base = I + (size_t)img * (1024u * 1024u) + (size_t)(4 * r + 1) * 1024u;
    float* dst = Rext + (size_t)row * REXTW;
    for (int col = threadIdx.x; col < REXTW; col += 256) {
        int s = col - RCOFF;
        float v = 0.f;
        if ((unsigned)s < (unsigned)PADW) {
            const float* p = base + (4 * reflect256(s - 30) + 1);
            v = 0.25f * (p[0] + p[1] + p[1024] + p[1025]);
        }
        dst[col] = v;
    }
}

// ---- zero the Hext row aprons (rows [0,33) and [349,384) per image) ------
__global__ void Pre_zero_apron(float* __restrict__ Hext) {
    int rr  = blockIdx.x;                     // 0..67
    int img = blockIdx.y;                     // 0..47
    int hrow = (rr < HROFF) ? rr : (HROFF + PADW + (rr - HROFF));  // 349 + ...
    Hext[((size_t)img * HEXTR + hrow) * OUTW + threadIdx.x] = 0.f;
}

// ---- horizontal FIR as banded GEMM (one wave = one 16x16 tile) -----------
// H[py,qx] = sum_s P[py,s] * f[s-qx+33]; A from Rext, B from fpad
__global__ __launch_bounds__(32) void Pre_hblur(const float* __restrict__ Rext,
                                                const float* __restrict__ fpad,
                                                float* __restrict__ Hext) {
    int tid = blockIdx.x;                     // 16 x-tiles * 20 y-tiles * 48 imgs
    int qt  = tid & 15;
    int t2  = tid >> 4;
    int yt  = t2 % 20;
    int img = t2 / 20;
    int qx0 = qt * 16;
    int py0 = yt * 16;

    int lane  = threadIdx.x;
    int mn    = lane & 15;                    // A row m == B col n
    int khalf = (lane >> 4) << 1;             // K {0,1} lanes 0-15, {2,3} lanes 16-31

    // A base: row = reflect(py0+m-30) of Rext; col = RCOFF + (qx0-33) + khalf (+4c+j)
    const float* ptrA = Rext + ((size_t)img * 256 + reflect256(py0 + mn - 30)) * REXTW
                             + (RCOFF - 33 + qx0 + khalf);          // 8B aligned
    // B base: fpad index (4c + khalf + j - n) + FOFF
    const float* fp = fpad + (FOFF + khalf - mn);

    v8f acc = {};
#pragma unroll
    for (int c = 0; c < 36; ++c) {
        v2f A = *(const v2f*)(ptrA + 4 * c);                 // global_load_b64
        v2f B = { fp[4 * c], fp[4 * c + 1] };                // two b32, L0-hot
        acc = __builtin_amdgcn_wmma_f32_16x16x4_f32(false, A, false, B,
                                                    (short)0, acc, false, false);
    }

    float* Hrow = Hext + ((size_t)img * HEXTR + (py0 + HROFF)) * OUTW + qx0 + mn;
    int mb = (lane >> 4) << 3;                // D: M = r + 8*(lane/16), N = lane&15
#pragma unroll
    for (int r = 0; r < 8; ++r) {
        int prow = py0 + mb + r;
        if (prow < PADW)
            Hrow[(size_t)(mb + r) * OUTW] = acc[r];
    }
}

// ---- vertical FIR as banded GEMM ----------------------------------------
// out[oy,ox] = sum_s f[s-oy+33] * H[s,ox]; A from fpad, B from Hext rows
__global__ __launch_bounds__(32) void Pre_vblur(const float* __restrict__ Hext,
                                                const float* __restrict__ fpad,
                                                float* __restrict__ out) {
    int tid = blockIdx.x;                     // 16 x-tiles * 16 y-tiles * 48 imgs
    int xt  = tid & 15;
    int t2  = tid >> 4;
    int yt  = t2 & 15;
    int img = t2 >> 4;
    int ox0 = xt * 16;
    int oy0 = yt * 16;

    int lane  = threadIdx.x;
    int mn    = lane & 15;
    int khalf = (lane >> 4) << 1;

    // A (filter): index 4c + khalf + j - m, tile-independent
    const float* fp = fpad + (FOFF + khalf - mn);
    // B (H rows): Hext row = s + 33 = oy0 + khalf + 4c + j; always in [0,384)
    const float* ptrB = Hext + ((size_t)img * HEXTR + (oy0 + khalf)) * OUTW + ox0 + mn;

    v8f acc = {};
#pragma unroll
    for (int c = 0; c < 36; ++c) {
        v2f A = { fp[4 * c], fp[4 * c + 1] };
        v2f B = { ptrB[(size_t)(4 * c) * OUTW], ptrB[(size_t)(4 * c + 1) * OUTW] };
        acc = __builtin_amdgcn_wmma_f32_16x16x4_f32(false, A, false, B,
                                                    (short)0, acc, false, false);
    }

    float* Orow = out + ((size_t)img * OUTW + oy0) * OUTW + ox0 + mn;
    int mb = (lane >> 4) << 3;
#pragma unroll
    for (int r = 0; r < 8; ++r) {
        Orow[(size_t)(mb + r) * OUTW] = acc[r];
    }
}

extern "C" void kernel_launch(void* const* d_in, const int* in_sizes, int n_in,
                              void* d_out, int out_size, void* d_ws, size_t ws_size,
                              hipStream_t stream) {
    const float* img = (const float*)d_in[0];        // [16,3,1024,1024] fp32
    float* out = (float*)d_out;                      // [16,3,256,256] fp32

    // ws layout: [0,1024) filter table; Rext 48*256*400 f32; Hext 48*384*256 f32
    float* fpad = (float*)d_ws;
    float* Rext = (float*)((char*)d_ws + 1024);
    float* Hext = (float*)((char*)d_ws + 1024 + (size_t)NIMG * 256 * REXTW * 4);

    Pre_init_filter<<<1, 192, 0, stream>>>(fpad);
    Pre_build_rext<<<NIMG * 256, 256, 0, stream>>>(img, Rext);
    Pre_zero_apron<<<dim3(HEXTR - PADW + 0 + 68 - (HEXTR - PADW), NIMG), 256, 0, stream>>>(Hext); // 68 apron rows
    Pre_hblur<<<16 * 20 * NIMG, 32, 0, stream>>>(Rext, fpad, Hext);
    Pre_vblur<<<16 * 16 * NIMG, 32, 0, stream>>>(Hext, fpad, out);
}